// YOLOLoss_28338194219069
// MI455X (gfx1250) — compile-verified
//
#include <hip/hip_runtime.h>
#include <math.h>

// ---------------------------------------------------------------------------
// YOLO loss for MI455X (gfx1250, wave32).
// pred: [B=1024, G=2704, A=3, 5] f32  (~166 MB -> bandwidth bound, ~7us floor
// at 23.3 TB/s). Strategy: coalesced b128 loads (4 slots = 80B = 5 float4),
// per-thread f32 accumulation, wave reduction via V_WMMA_F32_16X16X4_F32
// (full f32 precision: A = partials, B = ones -> D rows = row sums),
// LDS across waves, f32 global atomics. Tiny obj/finalize kernels follow.
// ---------------------------------------------------------------------------

#define GRID_HW 52
#define G_CELLS (GRID_HW * GRID_HW)      // 2704
#define N_ANCH 3
#define N_B 1024
#define INV_G (1.0f / 52.0f)
#define TOTAL_SLOTS ((unsigned)N_B * G_CELLS * N_ANCH)   // 8,306,688
#define TOTAL_GROUPS (TOTAL_SLOTS / 4u)                  // 2,076,672 (exact)

typedef float v2f __attribute__((ext_vector_type(2)));
typedef float v8f __attribute__((ext_vector_type(8)));

// Full-f32 wave32 sum using V_WMMA_F32_16X16X4_F32.
// A (16x4 f32, 2 VGPRs/lane): VGPR0 = lane partial (K=0 for lanes 0-15,
// K=2 for lanes 16-31), VGPR1 = 0.  B = all ones (layout independent).
// D[m][n] = rowsum(m); lane holds rows r (lanes 0-15) / r+8 (lanes 16-31)
// in its 8 D registers, so sum(d[0..7]) gives rows 0-7 or 8-15; two
// broadcasts finish the 32-lane total. Requires EXEC all-ones (call only
// from fully-converged block-level code).
__device__ __forceinline__ float wave_sum_wmma(float v) {
  v2f a; a[0] = v;    a[1] = 0.0f;
  v2f b; b[0] = 1.0f; b[1] = 1.0f;
  v8f c = {};
  v8f d = __builtin_amdgcn_wmma_f32_16x16x4_f32(
      /*neg_a=*/false, a, /*neg_b=*/false, b,
      /*c_mod=*/(short)0, c, /*reuse_a=*/false, /*reuse_b=*/false);
  float s = d[0] + d[1] + d[2] + d[3] + d[4] + d[5] + d[6] + d[7];
  return __shfl(s, 0, 32) + __shfl(s, 16, 32);
}

__device__ __forceinline__ float fsigmoid(float x) {
  return 1.0f / (1.0f + __expf(-x));
}

// ws layout (floats): 0=S_coor_all 1=S_conf_all 2=obj_coor_label
//                     3=obj_coor_ag 4=obj_conf_sq 5=obj_conf_iou
__global__ void yolo_init(float* __restrict__ ws) {
  if (threadIdx.x < 6) ws[threadIdx.x] = 0.0f;
}

// Main bandwidth-bound pass: decode all slots, accumulate the two global sums.
__global__ void yolo_noobj_reduce(const float* __restrict__ pred,
                                  const float* __restrict__ anchors,
                                  float* __restrict__ ws) {
  __shared__ float lds_coor[8];
  __shared__ float lds_conf[8];

  const unsigned tid     = blockIdx.x * blockDim.x + threadIdx.x;
  const unsigned gstride = gridDim.x * blockDim.x;

  // anchors is tiny & uniform -> scalar loads
  const float aw_[N_ANCH] = {anchors[0], anchors[2], anchors[4]};
  const float ah_[N_ANCH] = {anchors[1], anchors[3], anchors[5]};

  float accCoor = 0.0f, accConf = 0.0f;

  for (unsigned grp = tid; grp < TOTAL_GROUPS; grp += gstride) {
    const float4* pv = (const float4*)pred + (size_t)grp * 5;  // 80B, 16-aligned
    __builtin_prefetch((const void*)(pv + (size_t)gstride * 5), 0, 1);
    float4 q0 = pv[0], q1 = pv[1], q2 = pv[2], q3 = pv[3], q4 = pv[4];
    float e[20] = {q0.x, q0.y, q0.z, q0.w, q1.x, q1.y, q1.z, q1.w,
                   q2.x, q2.y, q2.z, q2.w, q3.x, q3.y, q3.z, q3.w,
                   q4.x, q4.y, q4.z, q4.w};

#pragma unroll
    for (int j = 0; j < 4; ++j) {
      unsigned s  = grp * 4u + (unsigned)j;
      unsigned a  = s % 3u;
      unsigned t  = s / 3u;
      unsigned g  = t % (unsigned)G_CELLS;
      float p0 = e[5 * j + 0], p1 = e[5 * j + 1], p2 = e[5 * j + 2];
      float p3 = e[5 * j + 3], p4 = e[5 * j + 4];

      float gx = (float)(g / GRID_HW);
      float gy = (float)(g % GRID_HW);
      float x  = (fsigmoid(p0) + gx) * INV_G;
      float y  = (fsigmoid(p1) + gy) * INV_G;
      float aw = aw_[a], ah = ah_[a];
      float w  = aw * __expf(p2);
      float h  = ah * __expf(p3);
      float cf = fsigmoid(p4);

      float agx = (gx + 0.5f) * INV_G;
      float agy = (gy + 0.5f) * INV_G;
      float dx = x - agx, dy = y - agy, dw = w - aw, dh = h - ah;
      accCoor += dx * dx + dy * dy + dw * dw + dh * dh;
      accConf += cf * cf;
    }
  }

  // wave32 reduction via WMMA (EXEC all-ones here: loop has reconverged)
  float wc = wave_sum_wmma(accCoor);
  float wf = wave_sum_wmma(accConf);

  unsigned lane = threadIdx.x & 31u;
  unsigned wid  = threadIdx.x >> 5;
  if (lane == 0) { lds_coor[wid] = wc; lds_conf[wid] = wf; }
  __syncthreads();
  if (threadIdx.x == 0) {
    float sc = 0.0f, sf = 0.0f;
    int nw = blockDim.x >> 5;
    for (int i = 0; i < nw; ++i) { sc += lds_coor[i]; sf += lds_conf[i]; }
    atomicAdd(&ws[0], sc);
    atomicAdd(&ws[1], sf);
  }
}

// Per-example object terms: responsible cell, anchor argmax, IoU.
__global__ void yolo_obj(const float* __restrict__ pred,
                         const float* __restrict__ label,
                         const float* __restrict__ anchors,
                         float* __restrict__ ws) {
  int b = blockIdx.x * blockDim.x + threadIdx.x;
  if (b >= N_B) return;

  float lx = label[b * 4 + 0], ly = label[b * 4 + 1];
  float lw = label[b * 4 + 2], lh = label[b * 4 + 3];

  int idx = (int)floorf(lx * 52.0f) * 52 + (int)floorf(ly * 52.0f);

  // argmax of squared distance (faithful to source); first max wins
  float best = -1.0f; int am = 0;
#pragma unroll
  for (int a = 0; a < N_ANCH; ++a) {
    float dw = lw - anchors[a * 2 + 0];
    float dh = lh - anchors[a * 2 + 1];
    float d  = dw * dw + dh * dh;
    if (d > best) { best = d; am = a; }
  }

  const float* p = pred + (((size_t)b * G_CELLS + idx) * N_ANCH + am) * 5;
  float gx = (float)(idx / GRID_HW), gy = (float)(idx % GRID_HW);
  float x  = (fsigmoid(p[0]) + gx) * INV_G;
  float y  = (fsigmoid(p[1]) + gy) * INV_G;
  float aw = anchors[am * 2 + 0], ah = anchors[am * 2 + 1];
  float w  = aw * __expf(p[2]);
  float h  = ah * __expf(p[3]);
  float cf = fsigmoid(p[4]);

  float agx = (gx + 0.5f) * INV_G, agy = (gy + 0.5f) * INV_G;

  float d0 = x - lx, d1 = y - ly, d2 = w - lw, d3 = h - lh;
  float coorLab = d0 * d0 + d1 * d1 + d2 * d2 + d3 * d3;
  float e0 = x - agx, e1 = y - agy, e2 = w - aw, e3 = h - ah;
  float coorAg = e0 * e0 + e1 * e1 + e2 * e2 + e3 * e3;

  // IoU (boxes clipped to [0,1]; 'areas' = x1*y1, faithful to source)
  float lx0 = fmaxf(lx - lw * 0.5f, 0.0f), ly0 = fmaxf(ly - lh * 0.5f, 0.0f);
  float lx1 = fminf(lx + lw * 0.5f, 1.0f), ly1 = fminf(ly + lh * 0.5f, 1.0f);
  float px0 = fmaxf(x - w * 0.5f, 0.0f),  py0 = fmaxf(y - h * 0.5f, 0.0f);
  float px1 = fminf(x + w * 0.5f, 1.0f),  py1 = fminf(y + h * 0.5f, 1.0f);
  float ix0 = fmaxf(lx0, px0), iy0 = fmaxf(ly0, py0);
  float ix1 = fminf(lx1, px1), iy1 = fminf(ly1, py1);
  float inter = fmaxf(ix1 - ix0, 0.0f) * fmaxf(iy1 - iy0, 0.0f);
  float la = lx1 * ly1, pa = px1 * py1;
  float iou = inter / (la + pa - inter);
  float dc  = cf - iou;

  atomicAdd(&ws[2], coorLab);
  atomicAdd(&ws[3], coorAg);
  atomicAdd(&ws[4], cf * cf);
  atomicAdd(&ws[5], dc * dc);
}

__global__ void yolo_finalize(const float* __restrict__ ws,
                              float* __restrict__ out) {
  if (threadIdx.x == 0 && blockIdx.x == 0) {
    const float nB = (float)N_B;
    const float n_noobj = nB * (float)(G_CELLS * N_ANCH - 1);  // 1024*8111
    float coor_obj   = ws[2] / (nB * 4.0f);
    float coor_noobj = (ws[0] - ws[3]) / (n_noobj * 4.0f);
    float conf_obj   = ws[5] / nB;
    float conf_noobj = (ws[1] - ws[4]) / n_noobj;
    out[0] = coor_obj + coor_noobj + conf_obj + conf_noobj;
  }
}

extern "C" void kernel_launch(void* const* d_in, const int* in_sizes, int n_in,
                              void* d_out, int out_size, void* d_ws, size_t ws_size,
                              hipStream_t stream) {
  const float* pred    = (const float*)d_in[0];  // [1024,2704,3,5] f32
  const float* label   = (const float*)d_in[1];  // [1024,4] f32
  const float* anchors = (const float*)d_in[2];  // [3,2] f32
  float* out = (float*)d_out;
  float* ws  = (float*)d_ws;

  (void)in_sizes; (void)n_in; (void)out_size; (void)ws_size;

  yolo_init<<<1, 32, 0, stream>>>(ws);
  yolo_noobj_reduce<<<2048, 256, 0, stream>>>(pred, anchors, ws);
  yolo_obj<<<(N_B + 255) / 256, 256, 0, stream>>>(pred, label, anchors, ws);
  yolo_finalize<<<1, 32, 0, stream>>>(ws, out);
}